// GaussianMoments3_60748017435176
// MI455X (gfx1250) — compile-verified
//
#include <hip/hip_runtime.h>
#include <hip/hip_bf16.h>

// -------------------- types --------------------
typedef __attribute__((ext_vector_type(2))) float v2f;
typedef __attribute__((ext_vector_type(8))) float v8f;

#define BATCH 512
#define KCL   64
#define DIM   64
#define DP    72        // LDS row pitch (floats): 2*DP % 64 == 16 -> conflict-free half-lane rows
#define NSLICE 8

// -------------------- transforms --------------------
__device__ __forceinline__ float sqrt_xform(float x) {
    // sign(sign(x)+0.1) * (sqrt(|x|+0.25) - 0.5); x==0 -> +1 branch
    float s = (x < 0.0f) ? -1.0f : 1.0f;
    return s * (__builtin_amdgcn_sqrtf(fabsf(x) + 0.25f) - 0.5f);
}
__device__ __forceinline__ float cbrt_xform(float x) {
    // sign(sign(x)+0.1) * ((|x|+0.19245...)^(1/3) - 0.57735...)
    float s = (x < 0.0f) ? -1.0f : 1.0f;
    float y = fabsf(x) + 0.19245008973f;  // strictly > 0
    // cbrt(y) = 2^(log2(y)/3): v_log_f32 + v_exp_f32 (2 hw transcendentals)
    float c = __builtin_amdgcn_exp2f(0.3333333333333f * __builtin_amdgcn_logf(y));
    return s * (c - 0.57735026919f);
}

// -------------------- kernel 1: argmax assignment + deterministic grouping --------------------
__global__ void gm3_setup_kernel(const float* __restrict__ logits,
                                 int* __restrict__ assign,
                                 int* __restrict__ counts,
                                 int* __restrict__ offsets,
                                 int* __restrict__ order) {
    int tid = threadIdx.x;  // 256 threads, single block
    for (int b = tid; b < BATCH; b += 256) {
        const float* row = logits + b * KCL;
        float best = row[0]; int bi = 0;
        #pragma unroll 4
        for (int j = 1; j < KCL; ++j) {
            float v = row[j];
            if (v > best) { best = v; bi = j; }   // first-max semantics (jnp.argmax)
        }
        assign[b] = bi;
    }
    __syncthreads();
    if (tid < KCL) {
        int c = 0;
        for (int b = 0; b < BATCH; ++b) c += (assign[b] == tid);
        counts[tid] = c;
    }
    __syncthreads();
    if (tid == 0) {
        int acc = 0;
        for (int k = 0; k < KCL; ++k) { offsets[k] = acc; acc += counts[k]; }
    }
    __syncthreads();
    if (tid < KCL) {
        int pos = offsets[tid];
        for (int b = 0; b < BATCH; ++b)
            if (assign[b] == tid) order[pos++] = b;   // b-ascending within cluster: deterministic
    }
}

// -------------------- kernel 2: cache cbrt_xform(gauss_moments3) (k-independent) --------------------
__global__ void gm3_t3_kernel(const float* __restrict__ g3, float* __restrict__ t3) {
    int i = blockIdx.x * blockDim.x + threadIdx.x;   // 262144 elements
    if (i < DIM * DIM * DIM) t3[i] = cbrt_xform(g3[i]);
}

// -------------------- kernel 3: per-cluster moments via WMMA, fused transform+reduce --------------------
__global__ void gm3_main_kernel(const float* __restrict__ emb,
                                const float* __restrict__ centers,
                                const float* __restrict__ w1,
                                const float* __restrict__ w2,
                                const float* __restrict__ w3,
                                const float* __restrict__ g1,
                                const float* __restrict__ g2,
                                const float* __restrict__ g3,
                                const float* __restrict__ t3,
                                const int* __restrict__ order,
                                const int* __restrict__ offsets,
                                const int* __restrict__ counts,
                                float* __restrict__ partials,
                                int use_t3) {
    extern __shared__ float sdh[];        // n_pad rows x DP floats (<= 512*72*4 = 144KB)
    __shared__ float red[256];

    const int k     = blockIdx.x;         // cluster
    const int slice = blockIdx.y;         // p3 tile slice
    const int tid   = threadIdx.x;
    const int n     = counts[k];
    const int off   = offsets[k];
    const int n_pad = (n + 3) & ~3;       // contraction padded to multiple of 4 (zero rows)

    // ---- stage masked diffs dh[i][c] = emb[order[off+i]][c] - centers[k][c] into LDS ----
    for (int idx = tid; idx < n_pad * DIM; idx += 256) {
        int i = idx >> 6, c = idx & 63;
        float v = 0.0f;
        if (i < n) v = emb[order[off + i] * DIM + c] - centers[k * DIM + c];
        sdh[i * DP + c] = v;
    }
    __syncthreads();

    const float inv = 1.0f / ((float)n + 1e-7f);
    const float cwn = (float)n * (1.0f / (float)BATCH);  // sum of counts == BATCH exactly

    // ---- p1 / p2 (small): only in slice 0 ----
    float p1 = 0.0f, p2 = 0.0f;
    if (slice == 0) {
        if (tid < DIM) {
            float s = 0.0f;
            for (int i = 0; i < n; ++i) s += sdh[i * DP + tid];
            float d1 = s * inv - g1[tid];
            p1 = w1[tid] * d1 * d1;
        }
        for (int p = tid; p < DIM * DIM; p += 256) {
            int d = p >> 6, e = p & 63;
            float s = 0.0f;
            for (int i = 0; i < n; ++i) s += sdh[i * DP + d] * sdh[i * DP + e];
            float dd = sqrt_xform(s * inv) - sqrt_xform(g2[p]);
            p2 += w2[p] * dd * dd;
        }
    }

    // ---- p3: moment3[k] = P^T * A GEMM, P[i,(d,e)] = dh_i[d]*dh_i[e], via V_WMMA_F32_16X16X4_F32 ----
    // Output C[(d,e), f]: 4096 x 64 -> 256 de-tiles x 4 f-tiles = 1024 16x16 tiles per cluster.
    const int wave = tid >> 5;
    const int lane = tid & 31;
    const int half = lane >> 4;     // K-split half of the fragment
    const int l    = lane & 15;
    const int ksteps = n_pad >> 2;

    float p3 = 0.0f;
    const int t_lo = slice * (1024 / NSLICE);
    const int t_hi = t_lo + (1024 / NSLICE);
    for (int t = t_lo + wave; t < t_hi; t += 8) {     // uniform trip count across waves
        const int de_t = t >> 2;
        const int f_t  = t & 3;
        const int d  = de_t >> 2;                      // fixed d within a de-tile
        const int e0 = (de_t & 3) << 4;
        const int f0 = f_t << 4;

        v8f c = {};
        for (int s = 0; s < ksteps; ++s) {
            const int r0 = (s << 2) + (half << 1);     // this half-lane covers K rows r0, r0+1
            const float* row0 = &sdh[r0 * DP];
            const float* row1 = &sdh[(r0 + 1) * DP];
            v2f a, b;
            // A (16x4, M=(d,e)-pair row, K=batch): A[m][kk] = dh[kk][d] * dh[kk][e0+m]
            a.x = row0[d] * row0[e0 + l];
            a.y = row1[d] * row1[e0 + l];
            // B (4x16, K=batch, N=f): B[kk][nn] = dh[kk][f0+nn]
            b.x = row0[f0 + l];
            b.y = row1[f0 + l];
            c = __builtin_amdgcn_wmma_f32_16x16x4_f32(
                    false, a, false, b, (short)0, c, false, false);
        }
        // epilogue: cbrt transform + weighted squared error, straight out of accumulators
        const int f = f0 + l;
        #pragma unroll
        for (int r = 0; r < 8; ++r) {
            const int m = r + (half << 3);             // C layout: VGPR r -> M = r / r+8
            const int e = e0 + m;
            const int def = ((d << 6) + e) * 64 + f;   // d*4096 + e*64 + f
            float tv = use_t3 ? t3[def] : cbrt_xform(g3[def]);
            float dd = cbrt_xform(c[r]) - tv;
            p3 += w3[def] * dd * dd;
        }
    }

    // ---- block reduction (deterministic tree) ----
    float total = cwn * (p1 + 0.5f * p2 + 0.25f * p3);
    red[tid] = total;
    __syncthreads();
    for (int s = 128; s > 0; s >>= 1) {
        if (tid < s) red[tid] += red[tid + s];
        __syncthreads();
    }
    if (tid == 0) partials[k * NSLICE + slice] = red[0];
}

// -------------------- kernel 4: deterministic final sum --------------------
__global__ void gm3_final_kernel(const float* __restrict__ partials, float* __restrict__ out, int np) {
    if (blockIdx.x == 0 && threadIdx.x == 0) {
        float s = 0.0f;
        for (int i = 0; i < np; ++i) s += partials[i];
        out[0] = s;
    }
}

// -------------------- host launch --------------------
extern "C" void kernel_launch(void* const* d_in, const int* in_sizes, int n_in,
                              void* d_out, int out_size, void* d_ws, size_t ws_size,
                              hipStream_t stream) {
    const float* emb     = (const float*)d_in[0];  // [512,64]
    const float* centers = (const float*)d_in[1];  // [64,64]
    const float* logits  = (const float*)d_in[2];  // [512,64]
    const float* w1      = (const float*)d_in[3];  // [64]
    const float* w2      = (const float*)d_in[4];  // [64,64]
    const float* w3      = (const float*)d_in[5];  // [64,64,64]
    const float* g1      = (const float*)d_in[6];  // [64]
    const float* g2      = (const float*)d_in[7];  // [64,64]
    const float* g3      = (const float*)d_in[8];  // [64,64,64]
    float* out = (float*)d_out;

    // workspace layout: small int region first so a tiny ws still works; t3 cache after.
    char* ws = (char*)d_ws;
    int*   assign   = (int*)(ws);                      // 512
    int*   counts   = assign + BATCH;                  // 64
    int*   offsets  = counts + KCL;                    // 64
    int*   order    = offsets + KCL;                   // 512
    float* partials = (float*)(order + BATCH);         // 64*NSLICE = 512
    const size_t t3_off = 8192;
    float* t3 = (float*)(ws + t3_off);                 // 262144 floats
    const int use_t3 = (ws_size >= t3_off + (size_t)DIM * DIM * DIM * sizeof(float)) ? 1 : 0;

    // allow 144KB dynamic LDS (gfx1250 WGP: up to 320KB per workgroup)
    const int lds_bytes = BATCH * DP * (int)sizeof(float);   // 147456
    (void)hipFuncSetAttribute(reinterpret_cast<const void*>(gm3_main_kernel),
                              hipFuncAttributeMaxDynamicSharedMemorySize, lds_bytes);

    gm3_setup_kernel<<<1, 256, 0, stream>>>(logits, assign, counts, offsets, order);
    if (use_t3) {
        gm3_t3_kernel<<<(DIM * DIM * DIM + 255) / 256, 256, 0, stream>>>(g3, t3);
    }
    gm3_main_kernel<<<dim3(KCL, NSLICE), 256, lds_bytes, stream>>>(
        emb, centers, w1, w2, w3, g1, g2, g3, t3, order, offsets, counts, partials, use_t3);
    gm3_final_kernel<<<1, 32, 0, stream>>>(partials, out, KCL * NSLICE);
}